// PillarVFE_20907900797227
// MI455X (gfx1250) — compile-verified
//
#include <hip/hip_runtime.h>

typedef __attribute__((ext_vector_type(2))) float v2f;
typedef __attribute__((ext_vector_type(4))) float v4f;
typedef __attribute__((ext_vector_type(8))) float v8f;

#define PCR0 (-51.2f)
#define PCR1 (-51.2f)
#define VS0  (0.1f)
#define VS1  (0.1f)
#define GRID_NX 1024
#define GRID_NY 1024
#define EPS  1e-5f
#define HSTR 72   // padded LDS row stride for 64-wide rows

// ---------------- zero workspace (4 floats: {zsum0,cnt0,zsum1,cnt1}) -------
__global__ void zero_ws_kernel(float* ws) {
    if (threadIdx.x < 4) ws[threadIdx.x] = 0.0f;
}

// ---------------- zero output (512 MB BEV grid, non-temporal) --------------
__global__ void zero_out_kernel(v4f* p, long n4) {
    const v4f zero = {0.f, 0.f, 0.f, 0.f};
    long stride = (long)gridDim.x * blockDim.x;
    for (long i = (long)blockIdx.x * blockDim.x + threadIdx.x; i < n4; i += stride)
        __builtin_nontemporal_store(zero, &p[i]);   // grid >> L2, don't pollute
}

// ---------------- z-mean reduction -----------------------------------------
__global__ void zmean_reduce_kernel(const float* __restrict__ pts,
                                    const unsigned char* __restrict__ msk,
                                    float* ws, int Npts) {
    float s0 = 0.f, c0 = 0.f, s1 = 0.f, c1 = 0.f;
    int total = 2 * Npts;
    int stride = gridDim.x * blockDim.x;
    for (int i = blockIdx.x * blockDim.x + threadIdx.x; i < total; i += stride) {
        float z = pts[(size_t)i * 5 + 2];
        float m = msk[i] ? 1.0f : 0.0f;
        if (i < Npts) { s0 += z * m; c0 += m; }
        else          { s1 += z * m; c1 += m; }
    }
    __shared__ float r[4][256];
    int t = threadIdx.x;
    r[0][t] = s0; r[1][t] = c0; r[2][t] = s1; r[3][t] = c1;
    __syncthreads();
    for (int off = 128; off > 0; off >>= 1) {
        if (t < off) {
            r[0][t] += r[0][t + off]; r[1][t] += r[1][t + off];
            r[2][t] += r[2][t + off]; r[3][t] += r[3][t + off];
        }
        __syncthreads();
    }
    if (t == 0) {
        atomicAdd(&ws[0], r[0][0]); atomicAdd(&ws[1], r[1][0]);
        atomicAdd(&ws[2], r[2][0]); atomicAdd(&ws[3], r[3][0]);
    }
}

// ---------------- main: featurize + PFN (WMMA f32) + scatter-max -----------
__global__ void __launch_bounds__(256)
pillar_vfe_kernel(const float* __restrict__ points,
                  const unsigned char* __restrict__ mask,
                  const float* __restrict__ W1,
                  const float* __restrict__ g1, const float* __restrict__ b1,
                  const float* __restrict__ m1, const float* __restrict__ v1,
                  const float* __restrict__ W2,
                  const float* __restrict__ g2, const float* __restrict__ b2,
                  const float* __restrict__ m2, const float* __restrict__ v2,
                  float* __restrict__ out, const float* __restrict__ ws,
                  int Npts, int tilesPerBatch, int totalTiles) {
    __shared__ float sW2t[64 * HSTR];   // W2 transposed: [c][k]
    __shared__ float sW1t[64 * 12];     // W1 transposed, K padded to 12: [c][kk]
    __shared__ float sS1[64], sT1[64], sS2[64], sT2[64];
    __shared__ float sH[8][16 * HSTR];  // per-wave h tile (16 points x 64 ch)
    __shared__ int   sIdx[8][16];
    __shared__ float sMsk[8][16];

    const int tid = threadIdx.x;
    // ---- stage weights / BN params into LDS (whole block) ----
    for (int i = tid; i < 64 * 64; i += 256) {
        int k = i >> 6, c = i & 63;
        sW2t[c * HSTR + k] = W2[i];          // W2[k][c] -> sW2t[c][k]
    }
    for (int i = tid; i < 64 * 12; i += 256) {
        int c = i / 12, kk = i % 12;
        sW1t[i] = (kk < 10) ? W1[kk * 64 + c] : 0.0f;
    }
    if (tid < 64) {
        float s = g1[tid] * rsqrtf(v1[tid] + EPS);
        sS1[tid] = s; sT1[tid] = b1[tid] - m1[tid] * s;
        float s2 = g2[tid] * rsqrtf(v2[tid] + EPS);
        sS2[tid] = s2; sT2[tid] = b2[tid] - m2[tid] * s2;
    }
    __syncthreads();

    const int wave = tid >> 5;
    const int lane = tid & 31;
    const int row  = lane & 15;       // M row of this lane
    const int hi   = lane >> 4;       // lane half selects K pair
    const int k0   = 2 * hi;
    const int col  = row;             // N column of this lane (B/C/D layouts)

    const int gw = blockIdx.x * 8 + wave;      // wave-uniform
    if (gw >= totalTiles) return;              // uniform branch, EXEC stays full

    const int b = gw / tilesPerBatch;
    const int t = gw - b * tilesPerBatch;
    int pi = t * 16 + row;
    const int inb = (pi < Npts);
    if (!inb) pi = Npts - 1;

    // ---- per-point aug features (fp32) ----
    const size_t pbase = ((size_t)b * Npts + pi) * 5;
    const float x = points[pbase + 0];
    const float y = points[pbase + 1];
    const float z = points[pbase + 2];
    const float inten = points[pbase + 3];
    const float ring  = points[pbase + 4];
    float m = (inb && mask[(size_t)b * Npts + pi]) ? 1.0f : 0.0f;

    const float zmean = ws[2 * b] / fmaxf(ws[2 * b + 1], 1.0f);

    int px = (int)floorf((x - PCR0) / VS0);
    int py = (int)floorf((y - PCR1) / VS1);
    px = min(max(px, 0), GRID_NX - 1);
    py = min(max(py, 0), GRID_NY - 1);
    const int idx = py * GRID_NX + px;

    const float cx = (float)px * VS0 + PCR0;
    const float cy = (float)py * VS1 + PCR1;
    const float x_c = x - (cx + VS0 * 0.5f);
    const float y_c = y - (cy + VS1 * 0.5f);
    const float z_c = z - zmean;
    const float x_p = x - cx;
    const float y_p = y - cy;

    if (hi == 0) { sIdx[wave][row] = idx; sMsk[wave][row] = m; }

    // ---- layer 1: (16x12) x (12x64) via V_WMMA_F32_16X16X4_F32 ----
    // A layout (32-bit A 16x4): lane row=M, VGPR pair holds K = k0..k0+1.
    // aug = {x,y,z,inten,ring,x_c,y_c,z_c,x_p,y_p,0,0}; k0 in {0,2} -> plain selects.
    v2f a0; a0.x = hi ? z    : x;      a0.y = hi ? inten : y;
    v2f a1; a1.x = hi ? y_c  : ring;   a1.y = hi ? z_c   : x_c;
    v2f a2; a2.x = hi ? 0.0f : x_p;    a2.y = hi ? 0.0f  : y_p;

#pragma unroll
    for (int j = 0; j < 4; ++j) {
        const int c = 16 * j + col;
        const float* w1c = &sW1t[c * 12];
        v2f bb0; bb0.x = w1c[k0];     bb0.y = w1c[k0 + 1];
        v2f bb1; bb1.x = w1c[4 + k0]; bb1.y = w1c[4 + k0 + 1];
        v2f bb2; bb2.x = w1c[8 + k0]; bb2.y = w1c[8 + k0 + 1];
        v8f acc = {0.f, 0.f, 0.f, 0.f, 0.f, 0.f, 0.f, 0.f};
        acc = __builtin_amdgcn_wmma_f32_16x16x4_f32(false, a0, false, bb0, (short)0, acc, false, false);
        acc = __builtin_amdgcn_wmma_f32_16x16x4_f32(false, a1, false, bb1, (short)0, acc, false, false);
        acc = __builtin_amdgcn_wmma_f32_16x16x4_f32(false, a2, false, bb2, (short)0, acc, false, false);
        const float s = sS1[c], tt = sT1[c];
#pragma unroll
        for (int v = 0; v < 8; ++v) {
            const int rm = v + 8 * hi;                     // D layout: VGPR v -> row v+8*hi
            sH[wave][rm * HSTR + c] = fmaxf(acc[v] * s + tt, 0.0f);
        }
    }

    // make this wave's LDS h-tile visible across lanes (CDNA5 split counter)
    asm volatile("s_wait_dscnt 0" ::: "memory");

    // ---- layer 2: (16x64) x (64x64), 16 K-steps per N tile ----
    const float* hrow = &sH[wave][row * HSTR];
#pragma unroll
    for (int j = 0; j < 4; ++j) {
        const int c = 16 * j + col;
        const float* w2c = &sW2t[c * HSTR];
        v8f acc = {0.f, 0.f, 0.f, 0.f, 0.f, 0.f, 0.f, 0.f};
#pragma unroll
        for (int k = 0; k < 16; ++k) {
            v2f a;  a.x  = hrow[4 * k + k0]; a.y  = hrow[4 * k + k0 + 1];
            v2f bb; bb.x = w2c[4 * k + k0];  bb.y = w2c[4 * k + k0 + 1];
            acc = __builtin_amdgcn_wmma_f32_16x16x4_f32(false, a, false, bb, (short)0, acc, false, false);
        }
        const float s = sS2[c], tt = sT2[c];
#pragma unroll
        for (int v = 0; v < 8; ++v) {
            const int rm = v + 8 * hi;
            const float val = fmaxf(acc[v] * s + tt, 0.0f) * sMsk[wave][rm];
            if (val > 0.0f) {
                const size_t off = ((size_t)(b * 64 + c) << 20) + (size_t)sIdx[wave][rm];
                atomicMax((int*)out + off, __float_as_int(val));  // valid: val>=0, grid init 0
            }
        }
    }
}

extern "C" void kernel_launch(void* const* d_in, const int* in_sizes, int n_in,
                              void* d_out, int out_size, void* d_ws, size_t ws_size,
                              hipStream_t stream) {
    const float*         points = (const float*)d_in[0];
    const unsigned char* pmask  = (const unsigned char*)d_in[1];
    const float* W1 = (const float*)d_in[2];
    const float* g1 = (const float*)d_in[3];
    const float* b1 = (const float*)d_in[4];
    const float* m1 = (const float*)d_in[5];
    const float* v1 = (const float*)d_in[6];
    const float* W2 = (const float*)d_in[7];
    const float* g2 = (const float*)d_in[8];
    const float* b2 = (const float*)d_in[9];
    const float* m2 = (const float*)d_in[10];
    const float* v2 = (const float*)d_in[11];
    float* out = (float*)d_out;
    float* ws  = (float*)d_ws;

    const int Npts = in_sizes[0] / (2 * 5);               // B=2, 5 floats/point
    const int tilesPerBatch = (Npts + 15) / 16;
    const int totalTiles = 2 * tilesPerBatch;
    const int mainBlocks = (totalTiles + 7) / 8;          // 8 waves per 256-thr block

    zero_ws_kernel<<<1, 64, 0, stream>>>(ws);
    zmean_reduce_kernel<<<512, 256, 0, stream>>>(points, pmask, ws, Npts);
    zero_out_kernel<<<4096, 256, 0, stream>>>((v4f*)out, (long)out_size / 4);
    pillar_vfe_kernel<<<mainBlocks, 256, 0, stream>>>(
        points, pmask, W1, g1, b1, m1, v1, W2, g2, b2, m2, v2,
        out, ws, Npts, tilesPerBatch, totalTiles);
}